// ScoreNet_19585050870227
// MI455X (gfx1250) — compile-verified
//
#include <hip/hip_runtime.h>
#include <hip/hip_bf16.h>

typedef float v2f __attribute__((ext_vector_type(2)));
typedef float v8f __attribute__((ext_vector_type(8)));

#define GLOBAL_AS __attribute__((address_space(1)))
#define LOCAL_AS  __attribute__((address_space(3)))

#if defined(__has_builtin)
#  if __has_builtin(__builtin_amdgcn_global_load_async_to_lds_b32)
#    define USE_ASYNC_LDS 1
#  endif
#endif

#if defined(USE_ASYNC_LDS)
static __device__ __forceinline__ void wait_async0() {
#  if __has_builtin(__builtin_amdgcn_s_wait_asynccnt)
  __builtin_amdgcn_s_wait_asynccnt(0);
#  else
  asm volatile("s_wait_asynccnt 0x0" ::: "memory");
#  endif
}
#endif

// ---- workspace layout (floats) ---- total 787712 floats ~= 3.0 MB
#define WS_F      0u
#define WS_A      262144u
#define WS_BP     524288u
#define WS_S1     786432u
#define WS_T1     786688u
#define WS_SUM2   786944u
#define WS_SSQ2   787072u
#define WS_S2     787200u
#define WS_T2     787328u
#define WS_SUM3   787456u
#define WS_SSQ3   787520u
#define WS_S3     787584u
#define WS_T3     787648u

// K0: f[b][n][d] = 0.5*(feats[b][1+2n][d] + feats[b][2+2n][d])
__global__ __launch_bounds__(256) void k_pairavg(const float* __restrict__ feats,
                                                 float* __restrict__ f) {
  int id = blockIdx.x * 256 + threadIdx.x;          // 262144 total
  int d = id & 255, n = (id >> 8) & 255, b = id >> 16;
  const float* p = feats + ((size_t)(b * 513 + 1 + 2 * n)) * 256 + d;
  f[id] = 0.5f * (p[0] + p[256]);
}

// K1: a[b][o][n] = sum_d W1[o][d]   * f[b][n][d]
//     bp[b][o][n]= sum_d W1[o][256+d]* f[b][n][d]
// One wave per 16x16 tile, FP32 WMMA 16x16x4, K=256 -> 64 chained WMMAs x2.
__global__ __launch_bounds__(32) void k_gemm1(const float* __restrict__ f,
                                              const float* __restrict__ W1,
                                              float* __restrict__ a,
                                              float* __restrict__ bp) {
  const int lane = threadIdx.x;
  const int half = lane >> 4, m = lane & 15;
  const int n0 = blockIdx.x * 16, o0 = blockIdx.y * 16, b = blockIdx.z;
  const float* frow  = f  + ((size_t)(b * 256 + n0 + m)) * 256;
  const float* w1row = W1 + (size_t)(o0 + m) * 512;
  v8f acca = {}; v8f accb = {};
#pragma unroll
  for (int t = 0; t < 64; ++t) {
    int c = 4 * t + 2 * half;                        // K index pair base
    float2 bv = *(const float2*)(frow + c);
    float2 av = *(const float2*)(w1row + c);
    float2 bw = *(const float2*)(w1row + 256 + c);
    v2f B; B.x = bv.x; B.y = bv.y;
    v2f Aa; Aa.x = av.x; Aa.y = av.y;
    v2f Ab; Ab.x = bw.x; Ab.y = bw.y;
    acca = __builtin_amdgcn_wmma_f32_16x16x4_f32(false, Aa, false, B, (short)0, acca, false, false);
    accb = __builtin_amdgcn_wmma_f32_16x16x4_f32(false, Ab, false, B, (short)0, accb, false, false);
  }
#pragma unroll
  for (int v = 0; v < 8; ++v) {
    int o = o0 + v + 8 * half;
    size_t idx = ((size_t)(b * 256 + o)) * 256 + n0 + m;
    a[idx]  = acca[v];
    bp[idx] = accb[v];
  }
}

// K2: exact BN1 stats from separability of h = a_i + bp_j + b1.
// One block per channel o.
__global__ __launch_bounds__(256) void k_bn1(const float* __restrict__ a,
                                             const float* __restrict__ bp,
                                             const float* __restrict__ b1,
                                             const float* __restrict__ g1,
                                             const float* __restrict__ be1,
                                             float* __restrict__ s1,
                                             float* __restrict__ t1) {
  __shared__ float4 red[256];
  const int o = blockIdx.x, t = threadIdx.x;
  float Sa = 0.f, Sa2 = 0.f, Sb = 0.f, Sb2 = 0.f, Cross = 0.f;
  for (int b = 0; b < 4; ++b) {
    float va = a [((size_t)(b * 256 + o)) * 256 + t];
    float vb = bp[((size_t)(b * 256 + o)) * 256 + t];
    red[t] = make_float4(va, va * va, vb, vb * vb);
    __syncthreads();
    for (int s = 128; s > 0; s >>= 1) {
      if (t < s) {
        float4 x = red[t], y = red[t + s];
        red[t] = make_float4(x.x + y.x, x.y + y.y, x.z + y.z, x.w + y.w);
      }
      __syncthreads();
    }
    if (t == 0) {
      float4 r = red[0];
      Sa += r.x; Sa2 += r.y; Sb += r.z; Sb2 += r.w;
      Cross += r.x * r.z;                            // rowsum_a[b] * rowsum_b[b]
    }
    __syncthreads();
  }
  if (t == 0) {
    float bb = b1[o];
    float mu  = (Sa + Sb) * (1.0f / 1024.0f) + bb;
    float eh2 = (Sa2 + Sb2) * (1.0f / 1024.0f) + bb * bb
              + 2.0f * Cross * (1.0f / 262144.0f)
              + 2.0f * bb * (Sa + Sb) * (1.0f / 1024.0f);
    float var = eh2 - mu * mu;
    float rsig = rsqrtf(var + 1e-5f);
    s1[o] = g1[o] * rsig;
    t1[o] = g1[o] * (bb - mu) * rsig + be1[o];       // h1 = relu(s1*(a+bp) + t1)
  }
}

__global__ void k_zero(float* __restrict__ ws) {
  int t = blockIdx.x * 256 + threadIdx.x;
  if (t < 256)      ws[WS_SUM2 + t] = 0.0f;          // sum2(128)+ssq2(128) contiguous
  else if (t < 384) ws[WS_SUM3 + (t - 256)] = 0.0f;  // sum3(64)+ssq3(64) contiguous
}

__global__ void k_fin(const float* __restrict__ sum, const float* __restrict__ ssq,
                      const float* __restrict__ g,   const float* __restrict__ be,
                      float* __restrict__ s, float* __restrict__ t, int nch) {
  int c = blockIdx.x * 64 + threadIdx.x;
  if (c >= nch) return;
  const float inv = 1.0f / 262144.0f;
  float mu  = sum[c] * inv;
  float var = ssq[c] * inv - mu * mu;
  float rsig = rsqrtf(var + 1e-5f);
  s[c] = g[c] * rsig;
  t[c] = be[c] - g[c] * mu * rsig;                   // h = relu(s*y + t), y includes bias
}

// Fused per-pixel MLP pass. One workgroup = (b, i, 128 j's); 8 waves.
// STAGE==2: compute y2, accumulate BN2 stats.
// STAGE==3: recompute y2->h2 (LDS), compute y3, accumulate BN3 stats.
// STAGE==4: full chain, write output.
template <int STAGE>
__global__ __launch_bounds__(256) void k_fused(
    const float* __restrict__ a,  const float* __restrict__ bp,
    const float* __restrict__ s1, const float* __restrict__ t1,
    const float* __restrict__ W2, const float* __restrict__ b2,
    float* __restrict__ sum2, float* __restrict__ ssq2,
    const float* __restrict__ s2, const float* __restrict__ t2,
    const float* __restrict__ W3, const float* __restrict__ b3,
    float* __restrict__ sum3, float* __restrict__ ssq3,
    const float* __restrict__ s3, const float* __restrict__ t3,
    const float* __restrict__ W4, const float* __restrict__ b4,
    float* __restrict__ out) {
  extern __shared__ float sm[];
  float* bpl = sm;               // [128][258]  (stride 258 -> conflict-free b64)
  float* uu  = sm + 33024;       // [256]  u[c] = s1[c]*a[b][c][i] + t1[c]
  float* scl = sm + 33280;       // [256]  s1[c]
  float* h2l = sm + 33536;       // [128][130]  (STAGE>=3)
  float* h3l = sm + 50176;       // [128][66]   (STAGE==4)

  const int tid  = threadIdx.x;
  const int lane = tid & 31, wave = tid >> 5;
  const int half = lane >> 4, m = lane & 15;
  const int j0 = blockIdx.x * 128;
  const int i  = blockIdx.y;
  const int b  = blockIdx.z;

  // bp slice, transposed scatter into padded LDS rows.
  // CDNA5 path: per-lane async global->LDS copies (ASYNCcnt), no VGPR round-trip.
#if defined(USE_ASYNC_LDS)
  for (int idx = tid; idx < 128 * 256; idx += 256) {
    int c = idx >> 7, jl = idx & 127;
    const float* src = bp + ((size_t)(b * 256 + c)) * 256 + j0 + jl;
    float* dst = bpl + jl * 258 + c;
    __builtin_amdgcn_global_load_async_to_lds_b32(
        (GLOBAL_AS int*)(uintptr_t)src,
        (LOCAL_AS int*)(uintptr_t)dst, 0, 0);
  }
#else
  for (int idx = tid; idx < 128 * 256; idx += 256) {
    int c = idx >> 7, jl = idx & 127;
    bpl[jl * 258 + c] = bp[((size_t)(b * 256 + c)) * 256 + j0 + jl];
  }
#endif
  {                                                   // per-channel affine for layer 1
    float s = s1[tid];
    uu[tid]  = fmaf(s, a[((size_t)(b * 256 + tid)) * 256 + i], t1[tid]);
    scl[tid] = s;
  }
#if defined(USE_ASYNC_LDS)
  wait_async0();
#endif
  __syncthreads();

  // ---- layer 2: y2 = W2 @ relu(s1*(a+bp)+t1) + b2 ; wave -> 16 channels ----
  const int ch0 = wave * 16;
  float2 aF[64];                                      // A-fragments in VGPRs, reused 8x
  {
    const float* w2row = W2 + (size_t)(ch0 + m) * 256;
#pragma unroll
    for (int t = 0; t < 64; ++t) aF[t] = *(const float2*)(w2row + 4 * t + 2 * half);
  }
  float b2v[8], s2v[8], t2v[8];
#pragma unroll
  for (int v = 0; v < 8; ++v) {
    int ch = ch0 + v + 8 * half;
    b2v[v] = b2[ch];
    if (STAGE >= 3) { s2v[v] = s2[ch]; t2v[v] = t2[ch]; }
  }
  float rs[8], rq[8];
#pragma unroll
  for (int v = 0; v < 8; ++v) { rs[v] = 0.f; rq[v] = 0.f; }

  for (int jt = 0; jt < 8; ++jt) {
    const int jl = jt * 16 + m;
    const float* bprow = bpl + jl * 258;
    v8f acc = {};
#pragma unroll
    for (int t = 0; t < 64; ++t) {
      int c = 4 * t + 2 * half;
      float2 bpv = *(const float2*)(bprow + c);
      float2 u2  = *(const float2*)(uu + c);
      float2 sv  = *(const float2*)(scl + c);
      v2f B; B.x = fmaxf(fmaf(sv.x, bpv.x, u2.x), 0.0f);
             B.y = fmaxf(fmaf(sv.y, bpv.y, u2.y), 0.0f);
      v2f A; A.x = aF[t].x; A.y = aF[t].y;
      acc = __builtin_amdgcn_wmma_f32_16x16x4_f32(false, A, false, B, (short)0, acc, false, false);
    }
#pragma unroll
    for (int v = 0; v < 8; ++v) {
      float y = acc[v] + b2v[v];
      if (STAGE == 2) { rs[v] += y; rq[v] += y * y; }
      else h2l[jl * 130 + (ch0 + v + 8 * half)] = fmaxf(fmaf(s2v[v], y, t2v[v]), 0.0f);
    }
  }

  if (STAGE == 2) {
#pragma unroll
    for (int v = 0; v < 8; ++v) {
      float s = rs[v], q = rq[v];
#pragma unroll
      for (int msk = 1; msk < 16; msk <<= 1) {
        s += __shfl_xor(s, msk, 32);
        q += __shfl_xor(q, msk, 32);
      }
      if (m == 0) {
        atomicAdd(&sum2[ch0 + v + 8 * half], s);
        atomicAdd(&ssq2[ch0 + v + 8 * half], q);
      }
    }
    return;
  }
  __syncthreads();

  // ---- layer 3: y3 = W3 @ h2 + b3 ; 4 ch-tiles x 8 j-tiles over 8 waves ----
  const int ch3 = (wave & 3) * 16;
  const int jtb = (wave >> 2) * 4;
  float2 aF3[32];
  {
    const float* w3row = W3 + (size_t)(ch3 + m) * 128;
#pragma unroll
    for (int t = 0; t < 32; ++t) aF3[t] = *(const float2*)(w3row + 4 * t + 2 * half);
  }
  float b3v[8], s3v[8], t3v[8];
#pragma unroll
  for (int v = 0; v < 8; ++v) {
    int ch = ch3 + v + 8 * half;
    b3v[v] = b3[ch];
    if (STAGE == 4) { s3v[v] = s3[ch]; t3v[v] = t3[ch]; }
  }
  float rs3[8], rq3[8];
#pragma unroll
  for (int v = 0; v < 8; ++v) { rs3[v] = 0.f; rq3[v] = 0.f; }

  for (int q = 0; q < 4; ++q) {
    const int jl = (jtb + q) * 16 + m;
    const float* h2row = h2l + jl * 130;
    v8f acc = {};
#pragma unroll
    for (int t = 0; t < 32; ++t) {
      int c = 4 * t + 2 * half;
      float2 hv = *(const float2*)(h2row + c);
      v2f B; B.x = hv.x; B.y = hv.y;
      v2f A; A.x = aF3[t].x; A.y = aF3[t].y;
      acc = __builtin_amdgcn_wmma_f32_16x16x4_f32(false, A, false, B, (short)0, acc, false, false);
    }
#pragma unroll
    for (int v = 0; v < 8; ++v) {
      float y = acc[v] + b3v[v];
      if (STAGE == 3) { rs3[v] += y; rq3[v] += y * y; }
      else h3l[jl * 66 + (ch3 + v + 8 * half)] = fmaxf(fmaf(s3v[v], y, t3v[v]), 0.0f);
    }
  }

  if (STAGE == 3) {
#pragma unroll
    for (int v = 0; v < 8; ++v) {
      float s = rs3[v], q = rq3[v];
#pragma unroll
      for (int msk = 1; msk < 16; msk <<= 1) {
        s += __shfl_xor(s, msk, 32);
        q += __shfl_xor(q, msk, 32);
      }
      if (m == 0) {
        atomicAdd(&sum3[ch3 + v + 8 * half], s);
        atomicAdd(&ssq3[ch3 + v + 8 * half], q);
      }
    }
    return;
  }
  __syncthreads();

  // ---- layer 4: out = W4 . h3 + b4 ----
  if (tid < 128) {
    const float* h3row = h3l + tid * 66;
    float accum = b4[0];
#pragma unroll
    for (int c = 0; c < 64; ++c) accum = fmaf(W4[c], h3row[c], accum);
    out[((size_t)(b * 256 + i)) * 256 + j0 + tid] = accum;
  }
}

extern "C" void kernel_launch(void* const* d_in, const int* in_sizes, int n_in,
                              void* d_out, int out_size, void* d_ws, size_t ws_size,
                              hipStream_t stream) {
  (void)in_sizes; (void)n_in; (void)out_size; (void)ws_size;
  const float* feats = (const float*)d_in[0];
  const float* W1  = (const float*)d_in[1];
  const float* b1  = (const float*)d_in[2];
  const float* g1  = (const float*)d_in[3];
  const float* be1 = (const float*)d_in[4];
  const float* W2  = (const float*)d_in[5];
  const float* b2  = (const float*)d_in[6];
  const float* g2  = (const float*)d_in[7];
  const float* be2 = (const float*)d_in[8];
  const float* W3  = (const float*)d_in[9];
  const float* b3  = (const float*)d_in[10];
  const float* g3  = (const float*)d_in[11];
  const float* be3 = (const float*)d_in[12];
  const float* W4  = (const float*)d_in[13];
  const float* b4  = (const float*)d_in[14];
  float* ws  = (float*)d_ws;
  float* out = (float*)d_out;

  float* f    = ws + WS_F;
  float* a    = ws + WS_A;
  float* bp   = ws + WS_BP;
  float* s1p  = ws + WS_S1;  float* t1p  = ws + WS_T1;
  float* sum2 = ws + WS_SUM2; float* ssq2 = ws + WS_SSQ2;
  float* s2p  = ws + WS_S2;  float* t2p  = ws + WS_T2;
  float* sum3 = ws + WS_SUM3; float* ssq3 = ws + WS_SSQ3;
  float* s3p  = ws + WS_S3;  float* t3p  = ws + WS_T3;

  const size_t lds2 = 33536u * 4u;                 // bpl + uu + scl
  const size_t lds3 = 50176u * 4u;                 // + h2l
  const size_t lds4 = 58624u * 4u;                 // + h3l

  k_pairavg<<<1024, 256, 0, stream>>>(feats, f);
  k_gemm1<<<dim3(16, 16, 4), 32, 0, stream>>>(f, W1, a, bp);
  k_bn1<<<256, 256, 0, stream>>>(a, bp, b1, g1, be1, s1p, t1p);
  k_zero<<<2, 256, 0, stream>>>(ws);
  k_fused<2><<<dim3(2, 256, 4), 256, lds2, stream>>>(a, bp, s1p, t1p, W2, b2, sum2, ssq2,
      s2p, t2p, W3, b3, sum3, ssq3, s3p, t3p, W4, b4, out);
  k_fin<<<2, 64, 0, stream>>>(sum2, ssq2, g2, be2, s2p, t2p, 128);
  k_fused<3><<<dim3(2, 256, 4), 256, lds3, stream>>>(a, bp, s1p, t1p, W2, b2, sum2, ssq2,
      s2p, t2p, W3, b3, sum3, ssq3, s3p, t3p, W4, b4, out);
  k_fin<<<1, 64, 0, stream>>>(sum3, ssq3, g3, be3, s3p, t3p, 64);
  k_fused<4><<<dim3(2, 256, 4), 256, lds4, stream>>>(a, bp, s1p, t1p, W2, b2, sum2, ssq2,
      s2p, t2p, W3, b3, sum3, ssq3, s3p, t3p, W4, b4, out);
}